// FunctionModel_42073499632055
// MI455X (gfx1250) — compile-verified
//
#include <hip/hip_runtime.h>
#include <stdint.h>

// TDM descriptor operand vectors (6-arg builtin form on this toolchain:
//   __builtin_amdgcn_tensor_load_to_lds(u32x4 g0, i32x8 g1, i32x4 g2, i32x4 g3, i32x8, i32 cpol))
typedef unsigned int u32x4 __attribute__((ext_vector_type(4)));
typedef int          i32x8 __attribute__((ext_vector_type(8)));
typedef int          i32x4 __attribute__((ext_vector_type(4)));

#define EMB_DIM    512
#define VOCAB_SZ   50257
#define TOK_PER_WG 16   // gather mode w/ 16-bit indices: up to 16 rows per TDM op

__global__ __launch_bounds__(32)
void emb_gather_tdm(const int* __restrict__ x,
                    const float* __restrict__ emb,
                    float* __restrict__ out)
{
    // 16 rows x 512 f32 = 32 KB staging tile in LDS (written by TDM load, read by TDM store)
    __shared__ float tile[TOK_PER_WG * EMB_DIM];

    const uint32_t wg   = blockIdx.x;
    const int      base = (int)wg * TOK_PER_WG;   // grid sized so every block is full

    // ---- wave-uniform index fetch + even-token masking (pure scalar, branch-free) ----
    const int* xb = x + base;
    uint32_t idx[TOK_PER_WG];
#pragma unroll
    for (int i = 0; i < TOK_PER_WG; ++i) {
        uint32_t t = (uint32_t)xb[i];
        idx[i] = t & (0u - (t & 1u));             // even token id -> row 0
    }

    const uint32_t lds_off = (uint32_t)(uintptr_t)&tile[0];  // low 32 bits = LDS byte offset

    // ================= gather-load descriptor: emb rows -> LDS =================
    const uint64_t src = (uint64_t)(uintptr_t)emb;
    u32x4 g0l;
    g0l.x = 0x80000001u;                          // count=1 | gather_mode(bit31) | 16-bit idx(bit30=0)
    g0l.y = lds_off;                              // lds_addr
    g0l.z = (uint32_t)src;                        // global_addr[31:0]
    g0l.w = (uint32_t)((src >> 32) & 0x01FFFFFFu) | 0x80000000u; // global_addr[56:32] | type=2

    i32x8 g1l;
    g1l[0] = (int)(2u << 16);                     // data_size = 4 bytes
    g1l[1] = (int)((uint32_t)EMB_DIM << 16);      // tensor_dim0 = 512  (dw1[31:16] = low16)
    g1l[2] = (int)(((uint32_t)VOCAB_SZ & 0xFFFFu) << 16); // tensor_dim1 = 50257 (dw2[31:16] = low16)
    g1l[3] = (int)((uint32_t)EMB_DIM << 16);      // tile_dim0 = 512  (dw3[31:16])
    g1l[4] = TOK_PER_WG;                          // tile_dim1 = #gather indices = 16
    g1l[5] = EMB_DIM;                             // tensor_dim0_stride = 512
    g1l[6] = 0;
    g1l[7] = 0;

    // groups 2/3: sixteen 16-bit row indices
    i32x4 g2l, g3l;
    g2l[0] = (int)(idx[0]  | (idx[1]  << 16));
    g2l[1] = (int)(idx[2]  | (idx[3]  << 16));
    g2l[2] = (int)(idx[4]  | (idx[5]  << 16));
    g2l[3] = (int)(idx[6]  | (idx[7]  << 16));
    g3l[0] = (int)(idx[8]  | (idx[9]  << 16));
    g3l[1] = (int)(idx[10] | (idx[11] << 16));
    g3l[2] = (int)(idx[12] | (idx[13] << 16));
    g3l[3] = (int)(idx[14] | (idx[15] << 16));

    // ================= linear store descriptor: LDS -> out =================
    const uint64_t dst   = (uint64_t)(uintptr_t)out + (uint64_t)base * (EMB_DIM * 4);
    const uint32_t nelem = TOK_PER_WG * EMB_DIM;  // 8192, fits 16-bit tile_dim0

    u32x4 g0s;
    g0s.x = 0x00000001u;                          // count=1, no gather
    g0s.y = lds_off;
    g0s.z = (uint32_t)dst;
    g0s.w = (uint32_t)((dst >> 32) & 0x01FFFFFFu) | 0x80000000u; // type=2

    i32x8 g1s;
    g1s[0] = (int)(2u << 16);                     // data_size = 4 bytes
    g1s[1] = (int)((nelem & 0xFFFFu) << 16);      // tensor_dim0 low16
    g1s[2] = (int)(1u << 16);                     // tensor_dim0 hi16 = 0 | tensor_dim1 = 1
    g1s[3] = (int)(nelem << 16);                  // tile_dim0 = 8192 (1-D tile)
    g1s[4] = 0;                                   // tile_dim1/2 unused
    g1s[5] = (int)nelem;                          // tensor_dim0_stride
    g1s[6] = 0;
    g1s[7] = 0;

    i32x4 gz4 = {0, 0, 0, 0};
    i32x8 gz8 = {0, 0, 0, 0, 0, 0, 0, 0};

    // ---- issue DMA: gather into LDS, drain load, stream out ----
    __builtin_amdgcn_tensor_load_to_lds(g0l, g1l, g2l, g3l, gz8, 0);
    __builtin_amdgcn_s_wait_tensorcnt(0);         // LDS tile valid
    __builtin_amdgcn_tensor_store_from_lds(g0s, g1s, gz4, gz4, gz8, 0);
    __builtin_amdgcn_s_wait_tensorcnt(0);         // drain before wave exit
}

extern "C" void kernel_launch(void* const* d_in, const int* in_sizes, int n_in,
                              void* d_out, int out_size, void* d_ws, size_t ws_size,
                              hipStream_t stream) {
    const int*   x   = (const int*)d_in[0];      // (8, 4096) int32 token ids
    const float* emb = (const float*)d_in[1];    // (50257, 512) f32 table
    float*       out = (float*)d_out;            // (8, 4096, 512) f32

    const int ntok   = in_sizes[0];              // 32768 (multiple of 16)
    const int blocks = ntok / TOK_PER_WG;        // 2048 full blocks

    emb_gather_tdm<<<blocks, 32, 0, stream>>>(x, emb, out);
}